// LSMP_39032662786093
// MI455X (gfx1250) — compile-verified
//
#include <hip/hip_runtime.h>

#define P_W 1.0f
#define U_W 0.5f

#define PS 130              // padded plane stride (floats)
#define NPLANE (128 * 128)  // elements per input plane
#define NOUT   (64 * 64)    // elements per output plane

typedef unsigned int v4u __attribute__((ext_vector_type(4)));
typedef int          v4i __attribute__((ext_vector_type(4)));
typedef int          v8i __attribute__((ext_vector_type(8)));
typedef float        v4f __attribute__((ext_vector_type(4)));

__device__ __forceinline__ float max4f(float a, float b, float c, float d) {
    return fmaxf(fmaxf(a, b), fmaxf(c, d));
}

__global__ void __launch_bounds__(256)
lift_kernel(const float* __restrict__ x, float* __restrict__ out) {
    // One whole zero-padded plane lives in LDS: 130*130*4 = 67,600 B.
    // gfx1250 WGP has 320 KB LDS -> 4 workgroups resident per WGP (32 wave32).
    __shared__ float tile[PS * PS];

    const int plane = blockIdx.x;
    const int tid   = threadIdx.x;
    const float* src = x + (size_t)plane * NPLANE;

    // ------------------------------------------------------------------
    // Issue one TDM DMA: 128x128 fp32 tile -> LDS interior at stride 130.
    // LDS-side padding: after every 128 dwords (pad_interval code 6) skip
    // 2 dwords (pad_amount code 1)  => LDS row stride = 130 floats.
    // Only wave 0 issues + waits TENSORcnt; barrier publishes to the WG.
    // ------------------------------------------------------------------
    if (tid < 32) {
        unsigned long long ga = (unsigned long long)(uintptr_t)src;
        // Low 32 bits of a flat LDS address are the LDS byte offset (CDNA5
        // aperture layout).
        unsigned int lds_addr =
            (unsigned int)(uintptr_t)(&tile[0]) + (PS + 1) * 4u;

        v4u g0;
        g0[0] = 1u;                                        // count=1 (user D#)
        g0[1] = lds_addr;                                  // lds_addr [63:32]
        g0[2] = (unsigned int)(ga & 0xFFFFFFFFu);          // global_addr lo
        g0[3] = (unsigned int)((ga >> 32) & 0x01FFFFFFu)   // global_addr hi
              | (2u << 30);                                // type = 2 (image)

        v8i g1;
        g1[0] = (2 << 16)      // data_size = 4 bytes
              | (1 << 20)      // pad_enable
              | (6 << 22)      // pad_interval: every 128 dwords
              | (1 << 25);     // pad_amount: 2 dwords
        g1[1] = (128 << 16);   // tensor_dim0 = 128 (bits 79:48 lo half)
        g1[2] = (128 << 16);   // tensor_dim1 = 128 (bits 111:80 lo half)
        g1[3] = (128 << 16);   // tile_dim0   = 128 (bits 127:112)
        g1[4] = 128;           // tile_dim1   = 128 (bits 143:128)
        g1[5] = 128;           // tensor_dim0_stride = 128 (bits 207:160 lo)
        g1[6] = 0;             // tensor_dim1_stride unused (2D)
        g1[7] = 0;

        v4i z4 = {0, 0, 0, 0};
#if __clang_major__ >= 23
        v8i z8 = {0, 0, 0, 0, 0, 0, 0, 0};
        __builtin_amdgcn_tensor_load_to_lds(g0, g1, z4, z4, z8, 0);
#else
        __builtin_amdgcn_tensor_load_to_lds(g0, g1, z4, z4, 0);
#endif
    }

    // Zero the 1-wide border while the TDM is in flight (disjoint LDS bytes).
    for (int k = tid; k < PS; k += 256) {
        tile[0 * PS + k]   = 0.0f;
        tile[129 * PS + k] = 0.0f;
        tile[k * PS + 0]   = 0.0f;
        tile[k * PS + 129] = 0.0f;
    }

    if (tid < 32) {
        __builtin_amdgcn_s_wait_tensorcnt(0);
    }
    __syncthreads();

    // ------------------------------------------------------------------
    // Pass A: steps 1 & 2 (predict). Targets (odd,even) and (even,odd);
    // both read only (odd,odd)/(even,even) -> fusable, race-free.
    // ------------------------------------------------------------------
    #pragma unroll
    for (int k = 0; k < 16; ++k) {
        int idx = tid + 256 * k;
        int i = idx >> 6, j = idx & 63;
        {   // step 1: (1+2i, 2+2j) -= max(up,left,down,right)
            int r = 1 + 2 * i, c = 2 + 2 * j;
            float m = max4f(tile[(r - 1) * PS + c], tile[r * PS + c - 1],
                            tile[(r + 1) * PS + c], tile[r * PS + c + 1]);
            tile[r * PS + c] -= P_W * m;
        }
        {   // step 2: (2+2i, 1+2j) -= max(up,left,down,right)
            int r = 2 + 2 * i, c = 1 + 2 * j;
            float m = max4f(tile[(r - 1) * PS + c], tile[r * PS + c - 1],
                            tile[(r + 1) * PS + c], tile[r * PS + c + 1]);
            tile[r * PS + c] -= P_W * m;
        }
    }
    __syncthreads();

    // ------------------------------------------------------------------
    // Pass B: step 3 (update odd,odd from pass-A results).
    // ------------------------------------------------------------------
    #pragma unroll
    for (int k = 0; k < 16; ++k) {
        int idx = tid + 256 * k;
        int i = idx >> 6, j = idx & 63;
        int r = 1 + 2 * i, c = 1 + 2 * j;
        float m = max4f(tile[(r - 1) * PS + c], tile[r * PS + c - 1],
                        tile[(r + 1) * PS + c], tile[r * PS + c + 1]);
        tile[r * PS + c] += U_W * m;
    }
    __syncthreads();

    // ------------------------------------------------------------------
    // Pass C: steps 4 & 5 fused (even,even += U_W*verhor - P_W*diag).
    // verhor reads pass-A sites; diag reads pass-B (odd,odd) sites;
    // neither reads (even,even) -> the two additive updates commute.
    // ------------------------------------------------------------------
    #pragma unroll
    for (int k = 0; k < 16; ++k) {
        int idx = tid + 256 * k;
        int i = idx >> 6, j = idx & 63;
        int r = 2 + 2 * i, c = 2 + 2 * j;
        float mv = max4f(tile[(r - 1) * PS + c], tile[r * PS + c - 1],
                         tile[(r + 1) * PS + c], tile[r * PS + c + 1]);
        float md = max4f(tile[(r - 1) * PS + c - 1], tile[(r + 1) * PS + c - 1],
                         tile[(r + 1) * PS + c + 1], tile[(r - 1) * PS + c + 1]);
        tile[r * PS + c] += U_W * mv - P_W * md;
    }
    __syncthreads();

    // ------------------------------------------------------------------
    // Pass D: step 6 fused with the output store. Result IS the (odd,odd)
    // quadrant, so no LDS write-back. Each thread produces 4 consecutive
    // output columns -> 16B nontemporal stores; 16 lanes cover a contiguous
    // 256 B row segment (coalesced). NT: output is write-once, never read,
    // keep it from displacing the streaming input in GL2.
    // ------------------------------------------------------------------
    float* dst = out + (size_t)plane * NOUT;
    const int tq   = tid & 15;   // 16 threads per output row
    const int rblk = tid >> 4;   // 16 rows concurrently
    #pragma unroll
    for (int it = 0; it < 4; ++it) {
        int i  = rblk + 16 * it;     // output row 0..63
        int j0 = 4 * tq;             // output col base 0,4,...,60
        int r  = 1 + 2 * i;
        v4f v;
        #pragma unroll
        for (int q = 0; q < 4; ++q) {
            int c = 1 + 2 * (j0 + q);
            float md = max4f(tile[(r - 1) * PS + c - 1], tile[(r + 1) * PS + c - 1],
                             tile[(r + 1) * PS + c + 1], tile[(r - 1) * PS + c + 1]);
            v[q] = tile[r * PS + c] + U_W * md;
        }
        __builtin_nontemporal_store(v, (v4f*)(dst + i * 64 + j0));
    }
}

extern "C" void kernel_launch(void* const* d_in, const int* in_sizes, int n_in,
                              void* d_out, int out_size, void* d_ws, size_t ws_size,
                              hipStream_t stream) {
    const float* x = (const float*)d_in[0];
    float* out = (float*)d_out;
    int planes = in_sizes[0] / NPLANE;   // 16 * 192 = 3072
    lift_kernel<<<dim3(planes), dim3(256), 0, stream>>>(x, out);
}